// QNN_Q3_54898271977537
// MI455X (gfx1250) — compile-verified
//
#include <hip/hip_runtime.h>

// ---------------------------------------------------------------------------
// 3-qubit StronglyEntanglingLayers QNN forward, mapped to CDNA5 wave32:
//   one wave per (b,t); one lane per 3-wire block (32 blocks == 32 lanes).
//   State: 8 complex amplitudes in registers. Inter-layer rewiring is pure
//   cross-lane traffic (ds_bpermute via __shfl on wave32).
//   Gate unitaries depend only on (t, layer, block) -> precomputed once and
//   reused by all B=128 batch elements (kills 18 of 21 sincos per circuit).
//   Depth-0 Rots act on a product state -> applied as per-qubit 2-vector
//   transforms + tensor product (~2x cheaper than full 8-state gates).
//   Trig: branchless 2-FMA Cody-Waite reduction + minimax polys.
// ---------------------------------------------------------------------------

__device__ __forceinline__ void fast_sincosf(float x, float* sp, float* cp) {
    float xn = rintf(x * 0.63661977236758134f);
    int q = (int)xn;
    float r = fmaf(xn, -1.57079637e+00f, x);   // -(pi/2)_hi, single-rounding via FMA
    r = fmaf(xn, 4.37113883e-08f, r);          // -(pi/2)_lo correction
    float r2 = r * r;
    float ss = fmaf(r2, 2.7557314e-6f, -1.9841271e-4f);   // sin, deg-9 minimax
    ss = fmaf(r2, ss, 8.3333310e-3f);
    ss = fmaf(r2, ss, -1.6666667e-1f);
    ss = fmaf(r * r2, ss, r);
    float cc = fmaf(r2, 2.4801587e-5f, -1.3888889e-3f);   // cos, deg-8 minimax
    cc = fmaf(r2, cc, 4.1666666e-2f);
    cc = fmaf(r2, cc, -0.5f);
    cc = fmaf(r2, cc, 1.0f);
    const int sw = q & 1;
    float s0 = sw ? cc : ss;
    float c0 = sw ? ss : cc;
    *sp = (q & 2) ? -s0 : s0;
    *cp = ((q + 1) & 2) ? -c0 : c0;
}

__device__ __forceinline__ void rot_params(float phi, float th, float om, float4& g) {
    // Rot = RZ(om) RY(th) RZ(phi):
    //   U00 = (e00,-f00), U01 = (-e01,-f01), U10 = (e01,-f01), U11 = (e00,f00)
    float st, ct, sp, cp, sm, cm;
    fast_sincosf(0.5f * th,         &st, &ct);
    fast_sincosf(0.5f * (phi + om), &sp, &cp);
    fast_sincosf(0.5f * (phi - om), &sm, &cm);
    g.x = cp * ct;  // e00
    g.y = sp * ct;  // f00
    g.z = cm * st;  // e01
    g.w = sm * st;  // f01
}

// Apply 2x2 complex unitary on the qubit whose index bit is BIT (full state).
template <int BIT>
__device__ __forceinline__ void apply1q(float ar[8], float ai[8], const float4 g) {
    const float e00 = g.x, f00 = g.y, e01 = g.z, f01 = g.w;
#pragma unroll
    for (int m0 = 0; m0 < 8; ++m0) {
        if (m0 & BIT) continue;
        const int m1 = m0 | BIT;
        const float p0r = ar[m0], p0i = ai[m0], p1r = ar[m1], p1i = ai[m1];
        ar[m0] = e00 * p0r + f00 * p0i - e01 * p1r + f01 * p1i;
        ai[m0] = e00 * p0i - f00 * p0r - e01 * p1i - f01 * p1r;
        ar[m1] = e01 * p0r + f01 * p0i + e00 * p1r - f00 * p1i;
        ai[m1] = e01 * p0i - f01 * p0r + e00 * p1i + f00 * p1r;
    }
}

// Apply 2x2 complex unitary to the real 2-vector (c,s) (single-qubit state).
__device__ __forceinline__ void rot_vec(const float4 g, float c, float s,
                                        float vr[2], float vi[2]) {
    vr[0] =  g.x * c - g.z * s;
    vi[0] = -g.y * c - g.w * s;
    vr[1] =  g.z * c + g.x * s;
    vi[1] = -g.w * c + g.y * s;
}

#define SWAP_AMP(a, b)                                   \
    do {                                                 \
        float _tr = ar[a]; ar[a] = ar[b]; ar[b] = _tr;   \
        float _ti = ai[a]; ai[a] = ai[b]; ai[b] = _ti;   \
    } while (0)

// Precompute gate parameters: one thread per Rot gate.
// theta flat: [T][3][32][2][3][3] ; gp flat: [T*3*32][6] float4
__global__ __launch_bounds__(256) void qnn_pre(const float* __restrict__ theta,
                                               float4* __restrict__ gp, int total) {
    int i = blockIdx.x * blockDim.x + threadIdx.x;
    if (i >= total) return;
    const float* w = theta + (size_t)i * 3;  // (circ*6 + rot)*3 == circ*18 + rot*3
    float4 g;
    rot_params(w[0], w[1], w[2], g);
    gp[i] = g;
}

template <bool PRECOMP>
__global__ __launch_bounds__(256) void qnn_main(const float* __restrict__ x,
                                                const float* __restrict__ wsrc,  // gp or theta
                                                float* __restrict__ out, int B, int T) {
    const int wid  = (int)((blockIdx.x * blockDim.x + threadIdx.x) >> 5);
    const int lane = threadIdx.x & 31;  // = block index within D (wave32)
    if (wid >= B * T) return;           // wave-uniform: EXEC stays all-ones for shuffles
    const int b = wid / T;
    const int t = wid % T;

    const float* xr = x + ((size_t)(b * T + t)) * 96 + lane * 3;
    float in0 = xr[0], in1 = xr[1], in2 = xr[2];

    float z0 = 0.f, z1 = 0.f, z2 = 0.f;

#pragma unroll
    for (int l = 0; l < 3; ++l) {
        const int circ = (t * 3 + l) * 32 + lane;

        float4 g[6];  // [depth][qubit] -> (e00,f00,e01,f01)
        if (PRECOMP) {
            const float4* gv = reinterpret_cast<const float4*>(wsrc) + (size_t)circ * 6;
#pragma unroll
            for (int j = 0; j < 6; ++j) g[j] = gv[j];  // global_load_b128 x6
        } else {
            const float* w = wsrc + (size_t)circ * 18;
#pragma unroll
            for (int j = 0; j < 6; ++j) rot_params(w[j * 3 + 0], w[j * 3 + 1], w[j * 3 + 2], g[j]);
        }

        // Embedding RY + depth-0 Rots act on a product state: do them per qubit.
        float c0, s0, c1, s1, c2, s2;
        fast_sincosf(0.5f * in0, &s0, &c0);
        fast_sincosf(0.5f * in1, &s1, &c1);
        fast_sincosf(0.5f * in2, &s2, &c2);

        float q0r[2], q0i[2], q1r[2], q1i[2], q2r[2], q2i[2];
        rot_vec(g[0], c0, s0, q0r, q0i);
        rot_vec(g[1], c1, s1, q1r, q1i);
        rot_vec(g[2], c2, s2, q2r, q2i);

        // Tensor product: amp[i,j,k] = q0[i] * q1[j] * q2[k]  (m = i*4 + j*2 + k)
        float ar[8], ai[8];
#pragma unroll
        for (int j = 0; j < 2; ++j) {
#pragma unroll
            for (int k = 0; k < 2; ++k) {
                const float tr = q1r[j] * q2r[k] - q1i[j] * q2i[k];
                const float ti = q1r[j] * q2i[k] + q1i[j] * q2r[k];
#pragma unroll
                for (int i = 0; i < 2; ++i) {
                    const int m = i * 4 + j * 2 + k;
                    ar[m] = q0r[i] * tr - q0i[i] * ti;
                    ai[m] = q0r[i] * ti + q0i[i] * tr;
                }
            }
        }

        // depth 0 CNOTs, range r=1: (0,1),(1,2),(2,0)
        SWAP_AMP(4, 6); SWAP_AMP(5, 7);  // CNOT(0,1)
        SWAP_AMP(2, 3); SWAP_AMP(6, 7);  // CNOT(1,2)
        SWAP_AMP(1, 5); SWAP_AMP(3, 7);  // CNOT(2,0)

        // depth 1: full-state Rots, then CNOT range r=2: (0,2),(1,0),(2,1)
        apply1q<4>(ar, ai, g[3]);
        apply1q<2>(ar, ai, g[4]);
        apply1q<1>(ar, ai, g[5]);
        SWAP_AMP(4, 5); SWAP_AMP(6, 7);  // CNOT(0,2)
        SWAP_AMP(2, 6); SWAP_AMP(3, 7);  // CNOT(1,0)
        SWAP_AMP(1, 3); SWAP_AMP(5, 7);  // CNOT(2,1)

        float p[8];
#pragma unroll
        for (int m = 0; m < 8; ++m) p[m] = ar[m] * ar[m] + ai[m] * ai[m];
        z0 = (p[0] + p[1] + p[2] + p[3]) - (p[4] + p[5] + p[6] + p[7]);
        z1 = (p[0] + p[1] + p[4] + p[5]) - (p[2] + p[3] + p[6] + p[7]);
        z2 = (p[0] + p[2] + p[4] + p[6]) - (p[1] + p[3] + p[5] + p[7]);

        if (l < 2) {
            // Rewiring: block i reads wires (3i-1, 3i+1, 3i+3) mod 96
            //  -> (lane-1).z2, (lane).z1, (lane+1).z0 : wave32 ds_bpermute
            in0 = __shfl(z2, (lane + 31) & 31, 32);
            in1 = z1;
            in2 = __shfl(z0, (lane + 1) & 31, 32);
        }
    }

    float* orow = out + ((size_t)(b * T + t)) * 96 + lane * 3;
    orow[0] = z0;
    orow[1] = z1;
    orow[2] = z2;
}

extern "C" void kernel_launch(void* const* d_in, const int* in_sizes, int n_in,
                              void* d_out, int out_size, void* d_ws, size_t ws_size,
                              hipStream_t stream) {
    const float* x     = (const float*)d_in[0];  // (B, T, 96) f32
    const float* theta = (const float*)d_in[1];  // (T, 3, 32, 2, 3, 3) f32
    float* out         = (float*)d_out;          // (B, T, 96) f32

    const int D = 96, NB = 32, NL = 3, LD = 2;
    const int T = in_sizes[1] / (NL * NB * LD * 9);  // 64
    const int B = in_sizes[0] / (T * D);             // 128

    const int nwaves = B * T;
    const int threads = 256;
    dim3 grd((unsigned)((nwaves * 32 + threads - 1) / threads));

    const int nrot = T * NL * NB * LD * 3;                 // total Rot gates (36864)
    const size_t gp_bytes = (size_t)nrot * sizeof(float4); // ~590 KB

    if (ws_size >= gp_bytes) {
        float4* gp = (float4*)d_ws;
        qnn_pre<<<dim3((unsigned)((nrot + threads - 1) / threads)), dim3(threads), 0, stream>>>(
            theta, gp, nrot);
        qnn_main<true><<<grd, dim3(threads), 0, stream>>>(x, (const float*)gp, out, B, T);
    } else {
        qnn_main<false><<<grd, dim3(threads), 0, stream>>>(x, theta, out, B, T);
    }
}